// HybridQueryGNN_12154757448074
// MI455X (gfx1250) — compile-verified
//
#include <hip/hip_runtime.h>
#include <hip/hip_bf16.h>
#include <math.h>

#define NEG_SLOPE 0.2f
#define EPSBN 1e-5f

typedef float v2f __attribute__((ext_vector_type(2)));
typedef float v4f __attribute__((ext_vector_type(4)));
typedef float v8f __attribute__((ext_vector_type(8)));

// ---- monotonic float<->uint mapping for atomic max on floats ----
__device__ __forceinline__ unsigned f2ord(float f) {
    unsigned u = __float_as_uint(f);
    return (u & 0x80000000u) ? ~u : (u | 0x80000000u);
}
__device__ __forceinline__ float ord2f(unsigned u) {
    u = (u & 0x80000000u) ? (u ^ 0x80000000u) : ~u;
    return __uint_as_float(u);
}

__global__ void fill_zero_u32(unsigned* __restrict__ p, long long n) {
    long long i = (long long)blockIdx.x * blockDim.x + threadIdx.x;
    if (i < n) p[i] = 0u;
}

// ============================================================
// Layer-1 projections: xl = x_nc @ Wl + bl ; xr = x_nc @ Wr + br
// x is [N,27]; x_nc drops column 24 (K = 26, padded to 28). Out: [N,64].
// One wave per 16x16 C tile. Unconditional clamped loads, value-masked
// K tail, 7x V_WMMA_F32_16X16X4_F32 chain, branch-free store fast path.
// ============================================================
__global__ void gemm_gat1(const float* __restrict__ x,
                          const float* __restrict__ Wl, const float* __restrict__ bl,
                          const float* __restrict__ Wr, const float* __restrict__ br,
                          float* __restrict__ xl, float* __restrict__ xr, int N) {
    const int lane = threadIdx.x & 31;
    const int mt = blockIdx.x;
    const int ct = blockIdx.y;                 // 0..7
    const bool right = ct >= 4;
    const float* __restrict__ W = right ? Wr : Wl;
    const float* __restrict__ bias = right ? br : bl;
    float* __restrict__ out = right ? xr : xl;
    const int ncol0 = (ct & 3) * 16;
    const int lo = lane & 15;
    const int hi = lane >> 4;
    const int m = mt * 16 + lo;
    const int mc = (m < N) ? m : (N - 1);      // clamped row (loads always legal)
    const int n = ncol0 + lo;
    const float* __restrict__ xrow = x + (long long)mc * 27;

    v2f a[7], b[7];
#pragma unroll
    for (int t = 0; t < 7; ++t) {
        const int k0 = 4 * t + 2 * hi;
        const int k1 = k0 + 1;
        // column remap (skip col 24) with clamp so addresses stay in range
        int i0 = (k0 < 24) ? k0 : (k0 + 1); i0 = (i0 < 27) ? i0 : 26;
        int i1 = (k1 < 24) ? k1 : (k1 + 1); i1 = (i1 < 27) ? i1 : 26;
        int w0 = (k0 < 26) ? k0 : 25;
        int w1 = (k1 < 26) ? k1 : 25;
        float ax = xrow[i0], ay = xrow[i1];
        float bx = W[w0 * 64 + n], by = W[w1 * 64 + n];
        a[t].x = (k0 < 26) ? ax : 0.f;
        a[t].y = (k1 < 26) ? ay : 0.f;
        b[t].x = (k0 < 26) ? bx : 0.f;
        b[t].y = (k1 < 26) ? by : 0.f;
    }
    v8f c = {0.f, 0.f, 0.f, 0.f, 0.f, 0.f, 0.f, 0.f};
#pragma unroll
    for (int t = 0; t < 7; ++t)
        c = __builtin_amdgcn_wmma_f32_16x16x4_f32(false, a[t], false, b[t], (short)0,
                                                  c, false, false);
    const float bv = bias[n];
    const int rbase = mt * 16 + 8 * hi;        // C layout: vgpr i -> row i + 8*hi
    float* __restrict__ outp = out + (long long)rbase * 64 + n;
    if (rbase + 8 <= N) {
#pragma unroll
        for (int i = 0; i < 8; ++i) outp[i * 64] = c[i] + bv;
    } else {
#pragma unroll
        for (int i = 0; i < 8; ++i)
            if (rbase + i < N) outp[i * 64] = c[i] + bv;
    }
}

// Layer-2 projections: [N,64] @ [64,32]. 16x WMMA chain; A loaded as b64 pairs.
__global__ void gemm_gat2(const float* __restrict__ h,
                          const float* __restrict__ Wl, const float* __restrict__ bl,
                          const float* __restrict__ Wr, const float* __restrict__ br,
                          float* __restrict__ xl, float* __restrict__ xr, int N) {
    const int lane = threadIdx.x & 31;
    const int mt = blockIdx.x;
    const int ct = blockIdx.y;                 // 0..3
    const bool right = ct >= 2;
    const float* __restrict__ W = right ? Wr : Wl;
    const float* __restrict__ bias = right ? br : bl;
    float* __restrict__ out = right ? xr : xl;
    const int ncol0 = (ct & 1) * 16;
    const int lo = lane & 15;
    const int hi = lane >> 4;
    const int m = mt * 16 + lo;
    const int mc = (m < N) ? m : (N - 1);
    const int n = ncol0 + lo;
    const v2f* __restrict__ hrow2 = (const v2f*)(h + (long long)mc * 64);  // 256B-aligned rows

    v2f a[16], b[16];
#pragma unroll
    for (int t = 0; t < 16; ++t) {
        const int k0 = 4 * t + 2 * hi;
        a[t] = hrow2[2 * t + hi];              // global_load_b64
        b[t].x = W[k0 * 32 + n];
        b[t].y = W[(k0 + 1) * 32 + n];
    }
    v8f c = {0.f, 0.f, 0.f, 0.f, 0.f, 0.f, 0.f, 0.f};
#pragma unroll
    for (int t = 0; t < 16; ++t)
        c = __builtin_amdgcn_wmma_f32_16x16x4_f32(false, a[t], false, b[t], (short)0,
                                                  c, false, false);
    const float bv = bias[n];
    const int rbase = mt * 16 + 8 * hi;
    float* __restrict__ outp = out + (long long)rbase * 32 + n;
    if (rbase + 8 <= N) {
#pragma unroll
        for (int i = 0; i < 8; ++i) outp[i * 32] = c[i] + bv;
    } else {
#pragma unroll
        for (int i = 0; i < 8; ++i)
            if (rbase + i < N) outp[i * 32] = c[i] + bv;
    }
}

// Per-edge GATv2 logits + segment max into nmax (ordered-uint atomic max).
// Row gathers vectorized to float4 (global_load_b128); rows are 128B-aligned.
__global__ void edge_logits(const float* __restrict__ xl, const float* __restrict__ xr,
                            const int* __restrict__ ei, const float* __restrict__ att,
                            float* __restrict__ logits, unsigned* __restrict__ nmax,
                            int E, int N, int H) {
    int e = blockIdx.x * blockDim.x + threadIdx.x;
    int Etot = E + N;
    if (e >= Etot) return;
    int s = (e < E) ? ei[e] : (e - E);
    int d = (e < E) ? ei[E + e] : (e - E);
    const int F = H * 32;
    for (int hh = 0; hh < H; ++hh) {
        const v4f* __restrict__ pl = (const v4f*)(xl + (long long)s * F + hh * 32);
        const v4f* __restrict__ pr = (const v4f*)(xr + (long long)d * F + hh * 32);
        const v4f* __restrict__ pa = (const v4f*)(att + hh * 32);
        float acc = 0.f;
#pragma unroll
        for (int q = 0; q < 8; ++q) {
            v4f vl = pl[q], vr = pr[q], va = pa[q];
#pragma unroll
            for (int k = 0; k < 4; ++k) {
                float v = vl[k] + vr[k];
                v = (v > 0.f) ? v : v * NEG_SLOPE;
                acc = fmaf(v, va[k], acc);
            }
        }
        logits[(long long)e * H + hh] = acc;
        atomicMax(&nmax[(long long)d * H + hh], f2ord(acc));
    }
}

// Per-edge: a = exp(logit - max); z[dst] += a; acc[dst] += a * xl[src].
__global__ void edge_accum(const float* __restrict__ xl, const int* __restrict__ ei,
                           const float* __restrict__ logits,
                           const unsigned* __restrict__ nmax,
                           float* __restrict__ z, float* __restrict__ acc,
                           int E, int N, int H) {
    int e = blockIdx.x * blockDim.x + threadIdx.x;
    int Etot = E + N;
    if (e >= Etot) return;
    int s = (e < E) ? ei[e] : (e - E);
    int d = (e < E) ? ei[E + e] : (e - E);
    const int F = H * 32;
    for (int hh = 0; hh < H; ++hh) {
        float mx = ord2f(nmax[(long long)d * H + hh]);
        float a = __expf(logits[(long long)e * H + hh] - mx);
        atomicAdd(&z[(long long)d * H + hh], a);
        const v4f* __restrict__ pl = (const v4f*)(xl + (long long)s * F + hh * 32);
        float* __restrict__ pd = acc + (long long)d * F + hh * 32;
#pragma unroll
        for (int q = 0; q < 8; ++q) {
            v4f vl = pl[q];
            atomicAdd(&pd[q * 4 + 0], a * vl[0]);
            atomicAdd(&pd[q * 4 + 1], a * vl[1]);
            atomicAdd(&pd[q * 4 + 2], a * vl[2]);
            atomicAdd(&pd[q * 4 + 3], a * vl[3]);
        }
    }
}

// Per-node: normalize by z and add bias -> pre-BN features.
__global__ void node_bias_div(const float* __restrict__ acc, const float* __restrict__ z,
                              const float* __restrict__ bias, float* __restrict__ hout,
                              int N, int H) {
    const int F = H * 32;
    long long idx = (long long)blockIdx.x * blockDim.x + threadIdx.x;
    if (idx >= (long long)N * F) return;
    int n = (int)(idx / F), j = (int)(idx % F);
    float zz = z[(long long)n * H + (j >> 5)];
    hout[idx] = acc[idx] / (zz + 1e-16f) + bias[j];
}

// One block per channel: mean and rsqrt(var + eps), no atomics.
__global__ void bn_stats(const float* __restrict__ h, int N, int C,
                         float* __restrict__ mu, float* __restrict__ istd) {
    __shared__ float ss[256], ss2[256];
    int j = blockIdx.x;
    float s = 0.f, s2 = 0.f;
    for (int n = threadIdx.x; n < N; n += blockDim.x) {
        float v = h[(long long)n * C + j];
        s += v;
        s2 = fmaf(v, v, s2);
    }
    ss[threadIdx.x] = s;
    ss2[threadIdx.x] = s2;
    __syncthreads();
    for (int st = blockDim.x / 2; st > 0; st >>= 1) {
        if ((int)threadIdx.x < st) {
            ss[threadIdx.x] += ss[threadIdx.x + st];
            ss2[threadIdx.x] += ss2[threadIdx.x + st];
        }
        __syncthreads();
    }
    if (threadIdx.x == 0) {
        float m = ss[0] / (float)N;
        float var = ss2[0] / (float)N - m * m;
        mu[j] = m;
        istd[j] = rsqrtf(var + EPSBN);
    }
}

__global__ void bn_elu(float* __restrict__ h, const float* __restrict__ mu,
                       const float* __restrict__ istd, const float* __restrict__ g,
                       const float* __restrict__ be, int N, int C) {
    long long idx = (long long)blockIdx.x * blockDim.x + threadIdx.x;
    if (idx >= (long long)N * C) return;
    int j = (int)(idx % C);
    float v = (h[idx] - mu[j]) * istd[j] * g[j] + be[j];
    h[idx] = (v > 0.f) ? v : expm1f(v);
}

// gate = relu(h2 @ Wg1 + bg1) @ Wg2 + bg2 ; segment max over graphs.
// Small MLP weights staged in LDS once per block.
__global__ void gate_kernel(const float* __restrict__ h2,
                            const float* __restrict__ Wg1, const float* __restrict__ bg1,
                            const float* __restrict__ Wg2, const float* __restrict__ bg2,
                            const int* __restrict__ batch, float* __restrict__ gate,
                            unsigned* __restrict__ gmax, int N) {
    __shared__ float sW[512], sb1[16], sW2[16];
    if (threadIdx.x < 16) {
        sb1[threadIdx.x] = bg1[threadIdx.x];
        sW2[threadIdx.x] = Wg2[threadIdx.x];
    }
    for (int i = threadIdx.x; i < 512; i += blockDim.x) sW[i] = Wg1[i];
    __syncthreads();

    int n = blockIdx.x * blockDim.x + threadIdx.x;
    if (n >= N) return;
    const v4f* __restrict__ hr4 = (const v4f*)(h2 + (long long)n * 32);
    float hr[32];
#pragma unroll
    for (int q = 0; q < 8; ++q) {
        v4f v = hr4[q];
        hr[q * 4 + 0] = v[0]; hr[q * 4 + 1] = v[1];
        hr[q * 4 + 2] = v[2]; hr[q * 4 + 3] = v[3];
    }
    float gsum = bg2[0];
#pragma unroll
    for (int j = 0; j < 16; ++j) {
        float t = sb1[j];
        for (int cc = 0; cc < 32; ++cc) t = fmaf(hr[cc], sW[cc * 16 + j], t);
        t = fmaxf(t, 0.f);
        gsum = fmaf(t, sW2[j], gsum);
    }
    gate[n] = gsum;
    atomicMax(&gmax[batch[n]], f2ord(gsum));
}

// Segment-softmax pooling + counts + root-cost sums.
__global__ void pool_kernel(const float* __restrict__ h2, const float* __restrict__ x,
                            const float* __restrict__ gate,
                            const unsigned* __restrict__ gmax,
                            const int* __restrict__ batch, float* __restrict__ gz,
                            float* __restrict__ hacc, float* __restrict__ cnt,
                            float* __restrict__ rc, int N) {
    int n = blockIdx.x * blockDim.x + threadIdx.x;
    if (n >= N) return;
    int b = batch[n];
    float a = __expf(gate[n] - ord2f(gmax[b]));
    atomicAdd(&gz[b], a);
    const v4f* __restrict__ hr4 = (const v4f*)(h2 + (long long)n * 32);
    float* __restrict__ pd = hacc + (long long)b * 32;
#pragma unroll
    for (int q = 0; q < 8; ++q) {
        v4f v = hr4[q];
        atomicAdd(&pd[q * 4 + 0], a * v[0]);
        atomicAdd(&pd[q * 4 + 1], a * v[1]);
        atomicAdd(&pd[q * 4 + 2], a * v[2]);
        atomicAdd(&pd[q * 4 + 3], a * v[3]);
    }
    atomicAdd(&cnt[b], 1.f);
    atomicAdd(&rc[b], x[(long long)n * 27 + 24]);
}

// Final head: per graph, [h_pooled(32) | root_cost] -> 16 -> 1.
__global__ void head_kernel(const float* __restrict__ hacc, const float* __restrict__ gz,
                            const float* __restrict__ cnt, const float* __restrict__ rc,
                            const float* __restrict__ Wh1, const float* __restrict__ bh1,
                            const float* __restrict__ Wh2, const float* __restrict__ bh2,
                            float* __restrict__ out, int G) {
    __shared__ float sW[33 * 16], sb1[16], sW2[16];
    if (threadIdx.x < 16) {
        sb1[threadIdx.x] = bh1[threadIdx.x];
        sW2[threadIdx.x] = Wh2[threadIdx.x];
    }
    for (int i = threadIdx.x; i < 33 * 16; i += blockDim.x) sW[i] = Wh1[i];
    __syncthreads();

    int g = blockIdx.x * blockDim.x + threadIdx.x;
    if (g >= G) return;
    float comb[33];
    float invz = 1.f / (gz[g] + 1e-16f);
#pragma unroll
    for (int cc = 0; cc < 32; ++cc) comb[cc] = hacc[(long long)g * 32 + cc] * invz;
    comb[32] = rc[g] / fmaxf(cnt[g], 1.f);
    float o = bh2[0];
#pragma unroll
    for (int j = 0; j < 16; ++j) {
        float t = sb1[j];
        for (int cc = 0; cc < 33; ++cc) t = fmaf(comb[cc], sW[cc * 16 + j], t);
        t = fmaxf(t, 0.f);
        o = fmaf(t, sW2[j], o);
    }
    out[g] = o;
}

extern "C" void kernel_launch(void* const* d_in, const int* in_sizes, int n_in,
                              void* d_out, int out_size, void* d_ws, size_t ws_size,
                              hipStream_t stream) {
    (void)n_in; (void)ws_size;
    const float* x    = (const float*)d_in[0];
    const int*   ei   = (const int*)d_in[1];
    const int*   batch= (const int*)d_in[2];
    const float* Wl1  = (const float*)d_in[3];
    const float* bl1  = (const float*)d_in[4];
    const float* Wr1  = (const float*)d_in[5];
    const float* br1  = (const float*)d_in[6];
    const float* att1 = (const float*)d_in[7];
    const float* b1   = (const float*)d_in[8];
    const float* g1   = (const float*)d_in[9];
    const float* be1  = (const float*)d_in[10];
    const float* Wl2  = (const float*)d_in[11];
    const float* bl2  = (const float*)d_in[12];
    const float* Wr2  = (const float*)d_in[13];
    const float* br2  = (const float*)d_in[14];
    const float* att2 = (const float*)d_in[15];
    const float* b2   = (const float*)d_in[16];
    const float* g2   = (const float*)d_in[17];
    const float* be2  = (const float*)d_in[18];
    const float* Wg1  = (const float*)d_in[19];
    const float* bg1  = (const float*)d_in[20];
    const float* Wg2  = (const float*)d_in[21];
    const float* bg2  = (const float*)d_in[22];
    const float* Wh1  = (const float*)d_in[23];
    const float* bh1  = (const float*)d_in[24];
    const float* Wh2  = (const float*)d_in[25];
    const float* bh2  = (const float*)d_in[26];

    const int N = in_sizes[0] / 27;
    const int E = in_sizes[1] / 2;
    const int Etot = E + N;
    const int G = out_size;

    // ---- workspace layout (floats), with phase-based reuse ----
    float* ws = (float*)d_ws;
    long long p = 0;
    float* xl1    = ws + p; p += (long long)N * 64;   // layer1 left proj; later acc2
    float* xr1    = ws + p; p += (long long)N * 64;   // layer1 right proj; later h2
    float* acc1   = ws + p; p += (long long)N * 64;   // layer1 msg acc; later xl2/xr2
    float* h1     = ws + p; p += (long long)N * 64;   // layer1 output
    float* logits = ws + p; p += (long long)Etot * 2; // edge logits; later gate[N]
    unsigned* nmax= (unsigned*)(ws + p); p += (long long)N * 2;
    float* z      = ws + p; p += (long long)N * 2;
    float* mu     = ws + p; p += 64;
    float* istd   = ws + p; p += 64;
    unsigned* gmax= (unsigned*)(ws + p); p += G;
    float* gz     = ws + p; p += G;
    float* hacc   = ws + p; p += (long long)G * 32;
    float* cnt    = ws + p; p += G;
    float* rc     = ws + p; p += G;

    float* xl2  = acc1;                        // [N,32]
    float* xr2  = acc1 + (long long)N * 32;    // [N,32]
    float* acc2 = xl1;                         // [N,32]
    float* h2   = xr1;                         // [N,32]
    float* gate = logits;                      // [N]

    auto zf = [&](void* ptr, long long nelem) {
        unsigned blocks = (unsigned)((nelem + 255) / 256);
        fill_zero_u32<<<dim3(blocks), dim3(256), 0, stream>>>((unsigned*)ptr, nelem);
    };

    const int mt = (N + 15) / 16;
    const int eb = (Etot + 255) / 256;
    const int nb = (N + 255) / 256;

    // ================= layer 1 =================
    zf(nmax, (long long)N * 2);
    zf(z, (long long)N * 2);
    zf(acc1, (long long)N * 64);
    gemm_gat1<<<dim3(mt, 8), dim3(32), 0, stream>>>(x, Wl1, bl1, Wr1, br1, xl1, xr1, N);
    edge_logits<<<dim3(eb), dim3(256), 0, stream>>>(xl1, xr1, ei, att1, logits, nmax, E, N, 2);
    edge_accum<<<dim3(eb), dim3(256), 0, stream>>>(xl1, ei, logits, nmax, z, acc1, E, N, 2);
    {
        long long tot = (long long)N * 64;
        node_bias_div<<<dim3((unsigned)((tot + 255) / 256)), dim3(256), 0, stream>>>(
            acc1, z, b1, h1, N, 2);
        bn_stats<<<dim3(64), dim3(256), 0, stream>>>(h1, N, 64, mu, istd);
        bn_elu<<<dim3((unsigned)((tot + 255) / 256)), dim3(256), 0, stream>>>(
            h1, mu, istd, g1, be1, N, 64);
    }

    // ================= layer 2 =================
    zf(nmax, N);
    zf(z, N);
    zf(acc2, (long long)N * 32);
    gemm_gat2<<<dim3(mt, 4), dim3(32), 0, stream>>>(h1, Wl2, bl2, Wr2, br2, xl2, xr2, N);
    edge_logits<<<dim3(eb), dim3(256), 0, stream>>>(xl2, xr2, ei, att2, logits, nmax, E, N, 1);
    edge_accum<<<dim3(eb), dim3(256), 0, stream>>>(xl2, ei, logits, nmax, z, acc2, E, N, 1);
    {
        long long tot = (long long)N * 32;
        node_bias_div<<<dim3((unsigned)((tot + 255) / 256)), dim3(256), 0, stream>>>(
            acc2, z, b2, h2, N, 1);
        bn_stats<<<dim3(32), dim3(256), 0, stream>>>(h2, N, 32, mu, istd);
        bn_elu<<<dim3((unsigned)((tot + 255) / 256)), dim3(256), 0, stream>>>(
            h2, mu, istd, g2, be2, N, 32);
    }

    // ================= pooling & head =================
    zf(gmax, G);
    zf(gz, G);
    zf(hacc, (long long)G * 32);
    zf(cnt, G);
    zf(rc, G);
    gate_kernel<<<dim3(nb), dim3(256), 0, stream>>>(h2, Wg1, bg1, Wg2, bg2, batch, gate, gmax, N);
    pool_kernel<<<dim3(nb), dim3(256), 0, stream>>>(h2, x, gate, gmax, batch, gz, hacc, cnt, rc, N);
    head_kernel<<<dim3((G + 63) / 64), dim3(64), 0, stream>>>(
        hacc, gz, cnt, rc, Wh1, bh1, Wh2, bh2, (float*)d_out, G);
}